// JKNet_65103114272768
// MI455X (gfx1250) — compile-verified
//
#include <hip/hip_runtime.h>
#include <hip/hip_bf16.h>
#include <math.h>

typedef __attribute__((ext_vector_type(2))) float v2f;
typedef __attribute__((ext_vector_type(8))) float v8f;

#define D_FEAT 132  /* LDS row stride (128 + 4 pad) to avoid bank conflicts */

// ---------------------------------------------------------------------------
// Degree accumulation: one thread per edge, float atomics into deg arrays.
// ---------------------------------------------------------------------------
__global__ void jk_degree_kernel(const int* __restrict__ src,
                                 const int* __restrict__ dst,
                                 float* __restrict__ deg_out,
                                 float* __restrict__ deg_in, int E) {
    int e = blockIdx.x * blockDim.x + threadIdx.x;
    if (e >= E) return;
    atomicAdd(&deg_out[src[e]], 1.0f);
    atomicAdd(&deg_in[dst[e]], 1.0f);
}

// In-place deg -> rsqrt(max(deg,1)) for both arrays (contiguous, 2N elems).
__global__ void jk_norm_kernel(float* __restrict__ deg, int n2) {
    int i = blockIdx.x * blockDim.x + threadIdx.x;
    if (i >= n2) return;
    float d = deg[i];
    deg[i] = rsqrtf(fmaxf(d, 1.0f));
}

// ---------------------------------------------------------------------------
// Init: out = in_feat (JK-max seed); xs = in_feat * norm_src (pre-scaled).
// One thread per float4 (row = tid/32).
// ---------------------------------------------------------------------------
__global__ void jk_init_kernel(const float* __restrict__ in_feat,
                               const float* __restrict__ norm_src,
                               float* __restrict__ out,
                               float* __restrict__ xs, int N) {
    int t = blockIdx.x * blockDim.x + threadIdx.x;
    int row = t >> 5;
    if (row >= N) return;
    int f = (t & 31) * 4;
    size_t idx = (size_t)row * 128 + f;
    float4 v = *(const float4*)(in_feat + idx);
    *(float4*)(out + idx) = v;
    float ns = norm_src[row];
    float4 s = make_float4(v.x * ns, v.y * ns, v.z * ns, v.w * ns);
    *(float4*)(xs + idx) = s;
}

// ---------------------------------------------------------------------------
// Aggregation: agg[dst] += xs[src]. One wave32 per edge; each lane owns a
// float4 slice of the 128 features -> broadcast index loads, 512B contiguous
// gathers, 4 global_atomic_add_f32 per lane.
// ---------------------------------------------------------------------------
__global__ void jk_agg_kernel(const float* __restrict__ xs,
                              const int* __restrict__ src,
                              const int* __restrict__ dst,
                              float* __restrict__ agg, int E) {
    int t = blockIdx.x * blockDim.x + threadIdx.x;
    int e = t >> 5;
    if (e >= E) return;
    int f = (t & 31) * 4;
    int s = src[e];
    int d = dst[e];
    float4 v = *(const float4*)(xs + (size_t)s * 128 + f);
    float* p = agg + (size_t)d * 128 + f;
    atomicAdd(p + 0, v.x);
    atomicAdd(p + 1, v.y);
    atomicAdd(p + 2, v.z);
    atomicAdd(p + 3, v.w);
}

// ---------------------------------------------------------------------------
// Fused GEMM + epilogue: one wave per 16-node tile.
//   A = (agg * norm_dst)[16 x 128]  (staged in LDS, stride 132)
//   B = W[128 x 128] row-major
//   C accumulated with v_wmma_f32_16x16x4_f32 (32 K-steps x 8 N-tiles)
// Epilogue: + bias, LeakyReLU(0.01), out = max(out, y), xs = y * norm_src.
// ---------------------------------------------------------------------------
__global__ __launch_bounds__(32) void jk_gemm_kernel(
    const float* __restrict__ agg, const float* __restrict__ W,
    const float* __restrict__ bias, const float* __restrict__ norm_dst,
    const float* __restrict__ norm_src, float* __restrict__ xs,
    float* __restrict__ out, int N) {
    __shared__ float tileA[16 * D_FEAT];

    const int lane = threadIdx.x;
    const int base = blockIdx.x * 16;

    // Stage scaled A tile: 512 float4 slots, 16 per lane.
#pragma unroll
    for (int i = 0; i < 16; ++i) {
        int idx = lane + i * 32;
        int row = idx >> 5;        // 0..15
        int c4  = (idx & 31) * 4;  // 0..124
        float4 v = make_float4(0.f, 0.f, 0.f, 0.f);
        float nd = 0.f;
        if (base + row < N) {
            v = *(const float4*)(agg + (size_t)(base + row) * 128 + c4);
            nd = norm_dst[base + row];
        }
        float4 sv = make_float4(v.x * nd, v.y * nd, v.z * nd, v.w * nd);
        *(float4*)(&tileA[row * D_FEAT + c4]) = sv;
    }
    __syncthreads();

    const int m    = lane & 15;
    const int koff = (lane >> 4) * 2;  // K pair selector per ISA A/B layout
    const int col0 = lane & 15;

    v8f acc[8] = {};

    for (int ks = 0; ks < 32; ++ks) {
        int k0 = ks * 4 + koff;
        v2f a;
        a.x = tileA[m * D_FEAT + k0];
        a.y = tileA[m * D_FEAT + k0 + 1];
        const float* wp = W + (size_t)k0 * 128 + col0;
#pragma unroll
        for (int nt = 0; nt < 8; ++nt) {
            v2f b;
            b.x = wp[nt * 16];        // W[k0][nt*16+col0]
            b.y = wp[nt * 16 + 128];  // W[k0+1][nt*16+col0]
            acc[nt] = __builtin_amdgcn_wmma_f32_16x16x4_f32(
                false, a, false, b, (short)0, acc[nt], false, false);
        }
    }

    // Epilogue. C layout: lane l, VGPR r -> (M=(l>>4)*8+r, N=l&15).
    const int mrow = (lane >> 4) * 8;
#pragma unroll
    for (int nt = 0; nt < 8; ++nt) {
        int col = nt * 16 + col0;
        float bv = bias[col];
#pragma unroll
        for (int r = 0; r < 8; ++r) {
            int row = base + mrow + r;
            if (row < N) {
                float h = acc[nt][r] + bv;
                float y = (h > 0.0f) ? h : 0.01f * h;
                size_t idx = (size_t)row * 128 + col;
                out[idx] = fmaxf(out[idx], y);
                xs[idx] = y * norm_src[row];
            }
        }
    }
}

// ---------------------------------------------------------------------------
extern "C" void kernel_launch(void* const* d_in, const int* in_sizes, int n_in,
                              void* d_out, int out_size, void* d_ws, size_t ws_size,
                              hipStream_t stream) {
    const float* in_feat = (const float*)d_in[0];
    const float* Ws      = (const float*)d_in[1];
    const float* bs      = (const float*)d_in[2];
    const int*   src     = (const int*)d_in[3];
    const int*   dst     = (const int*)d_in[4];
    float*       out     = (float*)d_out;

    const int D = 128;
    const int N = in_sizes[0] / D;
    const int L = in_sizes[1] / (D * D);
    const int E = in_sizes[3];

    float* norm_src = (float*)d_ws;        // N  (deg_out -> rsqrt in place)
    float* norm_dst = norm_src + N;        // N  (deg_in  -> rsqrt in place)
    float* xs       = norm_dst + N;        // N*D  pre-scaled features
    float* agg      = xs + (size_t)N * D;  // N*D  scatter-add target

    // Degrees + norms
    hipMemsetAsync(norm_src, 0, sizeof(float) * 2 * (size_t)N, stream);
    jk_degree_kernel<<<(E + 255) / 256, 256, 0, stream>>>(src, dst, norm_src,
                                                          norm_dst, E);
    jk_norm_kernel<<<(2 * N + 255) / 256, 256, 0, stream>>>(norm_src, 2 * N);

    // out = in_feat ; xs = in_feat * norm_src
    {
        long long threads = (long long)N * 32;
        jk_init_kernel<<<(unsigned)((threads + 255) / 256), 256, 0, stream>>>(
            in_feat, norm_src, out, xs, N);
    }

    const int mtiles = (N + 15) / 16;
    for (int l = 0; l < L; ++l) {
        hipMemsetAsync(agg, 0, sizeof(float) * (size_t)N * D, stream);
        {
            long long threads = (long long)E * 32;
            jk_agg_kernel<<<(unsigned)((threads + 255) / 256), 256, 0, stream>>>(
                xs, src, dst, agg, E);
        }
        jk_gemm_kernel<<<mtiles, 32, 0, stream>>>(
            agg, Ws + (size_t)l * D * D, bs + (size_t)l * D, norm_dst, norm_src,
            xs, out, N);
    }
}